// DNCMemory_61297773249162
// MI455X (gfx1250) — compile-verified
//
#include <hip/hip_runtime.h>

typedef __attribute__((ext_vector_type(2))) float v2f;
typedef __attribute__((ext_vector_type(4))) float v4f;
typedef __attribute__((ext_vector_type(8))) float v8f;
typedef __attribute__((ext_vector_type(4))) int   v4i;

#define B 32
#define H 4
#define S 4096
#define M 128
#define EPS 1e-8f

#define TSA 64            // s rows per block in dots kernel
#define TSC 128           // s rows per block in write/read kernel
#define TILESC (S / TSC)  // 32

// CDNA5 async global->LDS path (guarded: falls back to VGPR staging if the
// toolchain doesn't declare the builtins).
#if defined(__has_builtin)
#  if __has_builtin(__builtin_amdgcn_global_load_async_to_lds_b128) && \
      __has_builtin(__builtin_amdgcn_s_wait_asynccnt)
#    define USE_ASYNC_LDS 1
#  endif
#endif

#ifdef USE_ASYNC_LDS
__device__ __forceinline__ void async_g2l_b128(const float* gsrc, float* ldst) {
  __builtin_amdgcn_global_load_async_to_lds_b128(
      (__attribute__((address_space(1))) v4i*)gsrc,
      (__attribute__((address_space(3))) v4i*)ldst, 0, 0);
}
#endif

// ---------------------------------------------------------------------------
// Kernel A: cos[b,h,s] = dot(k[b,h,:], mem[b,s,:]) / (max(|k|,eps)*max(|m_s|,eps))
// Dots computed with V_WMMA_F32_16X16X4_F32 (fp32 matrix core path).
// ---------------------------------------------------------------------------
__global__ __launch_bounds__(128)
void kern_dots(const float* __restrict__ mem, const float* __restrict__ kk,
               float* __restrict__ cosbuf) {
  __shared__ float tile[TSA][132];  // padded stride to avoid LDS bank conflicts
  __shared__ float kl[H][132];
  __shared__ float mn[TSA];
  __shared__ float kn[H];

  const int t  = threadIdx.x;
  const int b  = blockIdx.y;
  const int s0 = blockIdx.x * TSA;

#ifdef USE_ASYNC_LDS
  // stage k[b] : H*M = 512 floats, one b128 per thread, coalesced
  {
    int h = t >> 5, c4 = (t & 31) * 4;
    async_g2l_b128(&kk[((size_t)b * H + h) * M + c4], &kl[h][c4]);
  }
  // stage memory tile: TSA x M = 8192 floats, 16 b128 per thread, coalesced
  for (int i = 0; i < 16; ++i) {
    int idx = t + i * 128;
    int row = idx >> 5, c4 = (idx & 31) * 4;
    async_g2l_b128(&mem[((size_t)b * S + s0 + row) * M + c4], &tile[row][c4]);
  }
  __builtin_amdgcn_s_wait_asynccnt(0);
#else
  {
    int h = t >> 5, c4 = (t & 31) * 4;
    v4f kv = *reinterpret_cast<const v4f*>(&kk[((size_t)b * H + h) * M + c4]);
    *reinterpret_cast<v4f*>(&kl[h][c4]) = kv;
  }
  for (int i = 0; i < 16; ++i) {
    int idx = t + i * 128;
    int row = idx >> 5, c4 = (idx & 31) * 4;
    v4f mv = *reinterpret_cast<const v4f*>(&mem[((size_t)b * S + s0 + row) * M + c4]);
    *reinterpret_cast<v4f*>(&tile[row][c4]) = mv;
  }
#endif
  __syncthreads();

  // per-row norms (2 threads per row, pair-combine via shfl), b128 LDS reads
  {
    int row = t >> 1, half = t & 1;
    float ss = 0.f;
    const v4f* rp = reinterpret_cast<const v4f*>(&tile[row][half * 64]);
    for (int i = 0; i < 16; ++i) {
      v4f v = rp[i];
      ss += v.x * v.x + v.y * v.y + v.z * v.z + v.w * v.w;
    }
    ss += __shfl_xor(ss, 1, 32);
    if (half == 0) mn[row] = fmaxf(sqrtf(ss), EPS);
  }
  if (t < 8) {
    int h = t >> 1, half = t & 1;
    float ss = 0.f;
    const v4f* rp = reinterpret_cast<const v4f*>(&kl[h][half * 64]);
    for (int i = 0; i < 16; ++i) {
      v4f v = rp[i];
      ss += v.x * v.x + v.y * v.y + v.z * v.z + v.w * v.w;
    }
    ss += __shfl_xor(ss, 1, 32);
    if (half == 0) kn[h] = fmaxf(sqrtf(ss), EPS);
  }
  __syncthreads();

  // WMMA: each wave computes a 16(s) x 16(col) tile, cols 0..3 = heads.
  // A (16x4 f32): lane L<16 -> row L, K={0,1}; lane>=16 -> row L-16, K={2,3}.
  const int wvid = t >> 5;           // wave 0..3 -> s rows [wvid*16, +16)
  const int lane = t & 31;
  const int lrow = lane & 15;        // A row within tile
  const int kb   = (lane >> 4) * 2;  // K sub-offset {0,2}
  const int hcol = lane & 15;        // output column (head)
  const int hc   = hcol < 4 ? hcol : 3;
  const bool hon = hcol < 4;
  const int r0   = wvid * 16;

  v8f c = {};
  for (int mk = 0; mk < M; mk += 4) {
    v2f a = *reinterpret_cast<const v2f*>(&tile[r0 + lrow][mk + kb]);
    v2f bb;
    float b0 = kl[hc][mk + kb];
    float b1 = kl[hc][mk + kb + 1];
    bb.x = hon ? b0 : 0.f;
    bb.y = hon ? b1 : 0.f;
    c = __builtin_amdgcn_wmma_f32_16x16x4_f32(false, a, false, bb,
                                              (short)0, c, false, false);
  }

  // D layout: lane<16 -> N=lane, VGPR r = M=r ; lane>=16 -> N=lane-16, M=8+r
  if (hon) {
    const int hi8 = (lane >> 4) * 8;
    const float kinv = 1.f / kn[hcol];
    const size_t base = ((size_t)b * H + hcol) * S + s0 + r0 + hi8;
#pragma unroll
    for (int r = 0; r < 8; ++r) {
      int sl = r0 + hi8 + r;
      cosbuf[base + r] = c[r] * (kinv / mn[sl]);
    }
  }
}

// ---------------------------------------------------------------------------
// Kernel B: w[b,h,:] = softmax(beta[b,h] * cos[b,h,:]) over S
// ---------------------------------------------------------------------------
__global__ __launch_bounds__(256)
void kern_softmax(const float* __restrict__ cosbuf, const float* __restrict__ beta,
                  float* __restrict__ wbuf) {
  __shared__ float red[256];
  const int t  = threadIdx.x;
  const int bh = blockIdx.x;
  const float bv = beta[bh];
  const size_t base = (size_t)bh * S;

  float z[16];
  float lm = -3.4e38f;
  for (int i = 0; i < 16; ++i) {
    z[i] = bv * cosbuf[base + t + i * 256];
    lm = fmaxf(lm, z[i]);
  }
  red[t] = lm; __syncthreads();
  for (int off = 128; off > 0; off >>= 1) {
    if (t < off) red[t] = fmaxf(red[t], red[t + off]);
    __syncthreads();
  }
  const float mx = red[0];
  __syncthreads();

  float ls = 0.f;
  for (int i = 0; i < 16; ++i) { z[i] = __expf(z[i] - mx); ls += z[i]; }
  red[t] = ls; __syncthreads();
  for (int off = 128; off > 0; off >>= 1) {
    if (t < off) red[t] += red[t + off];
    __syncthreads();
  }
  const float inv = 1.f / red[0];
  for (int i = 0; i < 16; ++i) wbuf[base + t + i * 256] = z[i] * inv;
}

// ---------------------------------------------------------------------------
// Kernel C: fused sequential erase/add + read partials per s-tile.
//   nmem never hits memory; per-block (H,M) partials written deterministically.
// ---------------------------------------------------------------------------
__global__ __launch_bounds__(256)
void kern_write_read(const float* __restrict__ mem, const float* __restrict__ wbuf,
                     const float* __restrict__ erase, const float* __restrict__ add,
                     float* __restrict__ part) {
  __shared__ float wl[H][TSC];
  __shared__ float el[H][M];
  __shared__ float al[H][M];
  __shared__ v4f red[256][H];

  const int t  = threadIdx.x;
  const int b  = blockIdx.y;
  const int s0 = blockIdx.x * TSC;

  for (int i = 0; i < 2; ++i) {
    int idx = t + i * 256; int h = idx >> 7; int s = idx & 127;
    wl[h][s] = wbuf[((size_t)b * H + h) * S + s0 + s];
  }
  for (int i = 0; i < 2; ++i) {
    int idx = t + i * 256; int h = idx >> 7; int m = idx & 127;
    el[h][m] = erase[((size_t)b * H + h) * M + m];
    al[h][m] = add[((size_t)b * H + h) * M + m];
  }
  __syncthreads();

  const int m4 = (t & 31) * 4;  // owned m columns
  const int rg = t >> 5;        // row group: rows rg, rg+8, ..., rg+120
  v4f ev[H], av[H], acc[H];
#pragma unroll
  for (int h = 0; h < H; ++h) {
    ev[h] = *reinterpret_cast<const v4f*>(&el[h][m4]);
    av[h] = *reinterpret_cast<const v4f*>(&al[h][m4]);
    v4f z = {0.f, 0.f, 0.f, 0.f};
    acc[h] = z;
  }

  for (int j = 0; j < 16; ++j) {
    const int sl = rg + j * 8;
    v4f nm = *reinterpret_cast<const v4f*>(&mem[((size_t)b * S + s0 + sl) * M + m4]);
#pragma unroll
    for (int h = 0; h < H; ++h) {      // sequential per-head, matches reference
      float wvh = wl[h][sl];
      nm = nm * (1.0f - wvh * ev[h]) + wvh * av[h];
    }
#pragma unroll
    for (int h = 0; h < H; ++h) acc[h] += wl[h][sl] * nm;
  }

#pragma unroll
  for (int h = 0; h < H; ++h) red[t][h] = acc[h];
  __syncthreads();

  if (t < 128) {  // reduce over the 8 row-groups, fixed order (deterministic)
    const int mg = t & 31, h = t >> 5;
    v4f ssum = red[mg][h];
    for (int r = 1; r < 8; ++r) ssum += red[r * 32 + mg][h];
    size_t o = (((size_t)blockIdx.x * B + b) * H + h) * M + mg * 4;
    *reinterpret_cast<v4f*>(&part[o]) = ssum;
  }
}

// ---------------------------------------------------------------------------
// Kernel D: reading[b,h,m] = sum over s-tiles of partials (fixed order)
// ---------------------------------------------------------------------------
__global__ __launch_bounds__(256)
void kern_reduce(const float* __restrict__ part, float* __restrict__ out) {
  const int o = blockIdx.x * 256 + threadIdx.x;  // < B*H*M
  float s = 0.f;
  for (int tix = 0; tix < TILESC; ++tix)
    s += part[(size_t)tix * (B * H * M) + o];
  out[o] = s;
}

// ---------------------------------------------------------------------------
extern "C" void kernel_launch(void* const* d_in, const int* in_sizes, int n_in,
                              void* d_out, int out_size, void* d_ws, size_t ws_size,
                              hipStream_t stream) {
  (void)in_sizes; (void)n_in; (void)out_size; (void)ws_size;
  const float* mem   = (const float*)d_in[0];
  const float* kk    = (const float*)d_in[1];
  const float* beta  = (const float*)d_in[2];
  const float* erase = (const float*)d_in[3];
  const float* add   = (const float*)d_in[4];
  float* out = (float*)d_out;

  float* wsf  = (float*)d_ws;
  float* cosb = wsf;                          // B*H*S
  float* wb   = wsf + (size_t)B * H * S;      // B*H*S
  float* part = wsf + 2 * (size_t)B * H * S;  // TILESC*B*H*M

  kern_dots<<<dim3(S / TSA, B), 128, 0, stream>>>(mem, kk, cosb);
  kern_softmax<<<dim3(B * H), 256, 0, stream>>>(cosb, beta, wb);
  kern_write_read<<<dim3(S / TSC, B), 256, 0, stream>>>(mem, wb, erase, add, part);
  kern_reduce<<<dim3((B * H * M) / 256), 256, 0, stream>>>(part, out);
}